// BrunelNetwork_1941325217858
// MI455X (gfx1250) — compile-verified
//
#include <hip/hip_runtime.h>
#include <hip/hip_bf16.h>
#include <stdint.h>

// ---- Brunel network constants (match reference) ----
#define N_TOT      10000
#define BATCH      8
#define T_STEPS    200
#define C_TOT      1000
#define RING_SLOTS 16            // 15-step delay via read slot (t+1)&15, write slot t&15
#define MASK_STRIDE 12288        // padded byte-mask per ring slot (multiple of 128*16*6)
#define V_TH       20.0f
#define V_RESET    10.0f
#define REF_CNT    20
#define DECAY      0.99501247919268232f  // exp(-DT/TAU_M) = exp(-0.005)

#define THREADS       128
#define NEUR_PER_BLK  32         // 32 neurons x 4 connection groups per block
#define GROUPS        4
#define I2_PER_GRP    125        // 250 connections per group as int2/float2 pairs
#define COPY_SWEEPS   6          // 128 threads * 16B * 6 = 12288 B mask stage

typedef int v4i __attribute__((ext_vector_type(4)));
typedef __attribute__((address_space(1))) v4i* gv4i_ptr;  // global v4i*
typedef __attribute__((address_space(3))) v4i* lv4i_ptr;  // LDS v4i*

// gfx1250 async global->LDS path (existence proven by round-1 diagnostics).
#if __has_builtin(__builtin_amdgcn_global_load_async_to_lds_b128) && \
    __has_builtin(__builtin_amdgcn_s_wait_asynccnt)
#define USE_ASYNC_LDS 1
#else
#define USE_ASYNC_LDS 0
#endif

__global__ __launch_bounds__(THREADS)
void brunel_step(const float* __restrict__ ext_t,        // [B,N] external input at t
                 const float* __restrict__ w,            // [N,C]
                 const int*   __restrict__ idx,          // [N,C]
                 const unsigned char* __restrict__ mask_rd,  // delayed spike bytes
                 unsigned char* __restrict__ mask_wr,        // new spike bytes
                 float* __restrict__ v_state,            // [B,N]
                 int*   __restrict__ ref_state,          // [B,N]
                 float* __restrict__ spk_out,            // [B,N] slice of d_out
                 float* __restrict__ v_out)              // [B,N] slice of d_out
{
    __shared__ __align__(16) unsigned int smask[MASK_STRIDE / 4];  // spike bytes, whole net
    __shared__ float red[NEUR_PER_BLK * BATCH];                    // cross-group partials

    const int tid = threadIdx.x;

    // ---- Stage delayed spike mask (1 byte/neuron, bit b = batch b) into LDS ----
#if USE_ASYNC_LDS
    {
        #pragma unroll
        for (int k = 0; k < COPY_SWEEPS; ++k) {
            const unsigned char* gsrc = mask_rd + k * (THREADS * 16) + tid * 16;
            const unsigned int   loff = (unsigned int)(uintptr_t)
                (((const unsigned char*)smask) + k * (THREADS * 16) + tid * 16);
            __builtin_amdgcn_global_load_async_to_lds_b128(
                (gv4i_ptr)(uintptr_t)gsrc, (lv4i_ptr)loff, 0, 0);
        }
        __builtin_amdgcn_s_wait_asynccnt(0);
    }
#else
    {
        const uint4* src = (const uint4*)mask_rd;
        uint4* dst = (uint4*)smask;
        #pragma unroll
        for (int k = 0; k < COPY_SWEEPS; ++k) dst[tid + THREADS * k] = src[tid + THREADS * k];
    }
#endif
    __syncthreads();

    // ---- Gather: each thread = (neuron, connection-group of 250) ----
    const int nsub = tid >> 2;           // 0..31
    const int grp  = tid & 3;            // 0..3
    const int n    = blockIdx.x * NEUR_PER_BLK + nsub;

    float acc[BATCH];
    #pragma unroll
    for (int b = 0; b < BATCH; ++b) acc[b] = 0.0f;

    if (n < N_TOT) {
        const int2*   idx2 = (const int2*)  (idx + (size_t)n * C_TOT) + grp * I2_PER_GRP;
        const float2* w2   = (const float2*)(w   + (size_t)n * C_TOT) + grp * I2_PER_GRP;
        const unsigned char* sm = (const unsigned char*)smask;

        for (int cb = 0; cb < I2_PER_GRP; cb += 25) {
            __builtin_prefetch((const void*)(idx2 + cb + 25), 0, 0);  // global_prefetch_b8
            __builtin_prefetch((const void*)(w2   + cb + 25), 0, 0);
            #pragma unroll 5
            for (int u = 0; u < 25; ++u) {
                const int2   j2 = idx2[cb + u];
                const float2 wv = w2[cb + u];
                const unsigned m0 = sm[j2.x];    // 8 batch bits in one LDS byte
                const unsigned m1 = sm[j2.y];
                #pragma unroll
                for (int b = 0; b < BATCH; ++b) {
                    acc[b] += (m0 & (1u << b)) ? wv.x : 0.0f;
                    acc[b] += (m1 & (1u << b)) ? wv.y : 0.0f;
                }
            }
        }
    }

    // ---- Reduce the 4 connection-groups via wave32 cross-lane ops ----
    #pragma unroll
    for (int b = 0; b < BATCH; ++b) {
        acc[b] += __shfl_xor(acc[b], 1, 32);
        acc[b] += __shfl_xor(acc[b], 2, 32);
    }
    if (grp == 0) {
        #pragma unroll
        for (int b = 0; b < BATCH; ++b) red[nsub * BATCH + b] = acc[b];
    }
    __syncthreads();

    // ---- LIF epilogue: threads 0..31 own one neuron each (coalesced I/O) ----
    if (tid < NEUR_PER_BLK) {
        const int nn = blockIdx.x * NEUR_PER_BLK + tid;
        if (nn < N_TOT) {
            unsigned spikebyte = 0u;
            #pragma unroll
            for (int b = 0; b < BATCH; ++b) {
                const int o = b * N_TOT + nn;
                const float i_rec = red[tid * BATCH + b];
                const float vprev = v_state[o];
                const int   r     = ref_state[o];
                float vn = vprev * DECAY + i_rec + ext_t[o];
                vn = (r > 0) ? V_RESET : vn;
                const bool  s  = (vn >= V_TH);
                const float vo = s ? V_RESET : vn;
                const int   ro = s ? REF_CNT : ((r > 0) ? (r - 1) : 0);
                spk_out[o]   = s ? 1.0f : 0.0f;
                v_out[o]     = vo;
                v_state[o]   = vo;
                ref_state[o] = ro;
                spikebyte |= (s ? 1u : 0u) << b;
            }
            mask_wr[nn] = (unsigned char)spikebyte;
        }
    }
}

__global__ void brunel_init(unsigned int* __restrict__ ring_u32,
                            float* __restrict__ v_state,
                            int* __restrict__ ref_state) {
    const int i = blockIdx.x * blockDim.x + threadIdx.x;
    const int stride = gridDim.x * blockDim.x;
    for (int k = i; k < (RING_SLOTS * MASK_STRIDE) / 4; k += stride) ring_u32[k] = 0u;
    for (int k = i; k < BATCH * N_TOT; k += stride) { v_state[k] = 0.0f; ref_state[k] = 0; }
}

extern "C" void kernel_launch(void* const* d_in, const int* in_sizes, int n_in,
                              void* d_out, int out_size, void* d_ws, size_t ws_size,
                              hipStream_t stream) {
    const float* ext = (const float*)d_in[0];   // [T,B,N]
    const float* w   = (const float*)d_in[1];   // [N,C]
    const int*   idx = (const int*)d_in[2];     // [N,C]

    float* spikes = (float*)d_out;                                   // [T,B,N]
    float* vtr    = (float*)d_out + (size_t)T_STEPS * BATCH * N_TOT; // [T,B,N]

    unsigned char* ring = (unsigned char*)d_ws;                      // 16 * 12288 B
    float* v_state = (float*)((char*)d_ws + RING_SLOTS * MASK_STRIDE);
    int*   ref_state = (int*)((char*)d_ws + RING_SLOTS * MASK_STRIDE
                              + (size_t)BATCH * N_TOT * sizeof(float));

    brunel_init<<<256, 256, 0, stream>>>((unsigned int*)ring, v_state, ref_state);

    const int nblk = (N_TOT + NEUR_PER_BLK - 1) / NEUR_PER_BLK;      // 313
    for (int t = 0; t < T_STEPS; ++t) {
        const unsigned char* mrd = ring + ((size_t)((t + 1) & 15)) * MASK_STRIDE;
        unsigned char*       mwr = ring + ((size_t)(t & 15)) * MASK_STRIDE;
        brunel_step<<<nblk, THREADS, 0, stream>>>(
            ext + (size_t)t * BATCH * N_TOT, w, idx, mrd, mwr,
            v_state, ref_state,
            spikes + (size_t)t * BATCH * N_TOT,
            vtr    + (size_t)t * BATCH * N_TOT);
    }
}